// RegionOfInterestPoolingLayer_2310692405697
// MI455X (gfx1250) — compile-verified
//
#include <hip/hip_runtime.h>
#include <cstdint>

// ROI bilinear pooling, MI455X (gfx1250).
// Memory-bound gather (AI ~0.25 flop/byte): no WMMA structure; optimize the
// CDNA5 data path instead: async global->LDS DMA (ASYNCcnt), 128-bit
// coalesced per-lane traffic, non-temporal streaming stores.
// 3-D grid (pw, ph, roi) avoids integer-division block decomposition (SALU).

typedef float v4f __attribute__((ext_vector_type(4)));

constexpr int POOL  = 7;
constexpr int N_ROI = 300;
constexpr int IMG_W = 200;
constexpr int CH    = 512;

__global__ __launch_bounds__(128)
void roi_pool_bilinear_async(const float* __restrict__ img,
                             const float* __restrict__ rois,
                             float* __restrict__ out)
{
    // 8 KB staging buffer: taps p00, p01, p10, p11 (512 floats each)
    __shared__ float taps[4 * CH];

    const int pw  = blockIdx.x;                  // 0..6
    const int ph  = blockIdx.y;                  // 0..6
    const int roi = blockIdx.z;                  // 0..299
    const int t   = threadIdx.x;                 // 0..127, 4 channels each

    // ROI params (uniform per block -> scalar loads)
    const float4 rv = *(const float4*)(rois + (size_t)roi * 4);
    const int rx = (int)rv.x, ry = (int)rv.y, rw = (int)rv.z, rh = (int)rv.w;

    // Legacy-resize source coords: s = p * (dim/7); i0=floor; i1=min(i0+1,dim-1)
    // NOTE: true f32 divide to match the reference's rounding exactly.
    const float sy = (float)ph * ((float)rh / (float)POOL);
    const float sx = (float)pw * ((float)rw / (float)POOL);
    const int   y0 = (int)sy;                    // sy >= 0 -> trunc == floor
    const int   x0 = (int)sx;
    const int   y1 = min(y0 + 1, rh - 1);
    const int   x1 = min(x0 + 1, rw - 1);
    const float fy = sy - (float)y0;
    const float fx = sx - (float)x0;

    const unsigned row0 = (unsigned)(ry + y0), row1 = (unsigned)(ry + y1);
    const unsigned col0 = (unsigned)(rx + x0), col1 = (unsigned)(rx + x1);
    const unsigned co   = (unsigned)t * 4u;      // channel offset (16B aligned)

    const float* g00 = img + ((size_t)(row0 * IMG_W + col0)) * CH + co;
    const float* g01 = img + ((size_t)(row0 * IMG_W + col1)) * CH + co;
    const float* g10 = img + ((size_t)(row1 * IMG_W + col0)) * CH + co;
    const float* g11 = img + ((size_t)(row1 * IMG_W + col1)) * CH + co;

    // LDS byte offsets (low 32 bits of generic pointer into LDS == AS3 offset)
    const unsigned l00 = (unsigned)(size_t)&taps[0 * CH + co];
    const unsigned l01 = (unsigned)(size_t)&taps[1 * CH + co];
    const unsigned l10 = (unsigned)(size_t)&taps[2 * CH + co];
    const unsigned l11 = (unsigned)(size_t)&taps[3 * CH + co];

    // CDNA5 async DMA: global -> LDS, tracked by ASYNCcnt (not LOADcnt).
    asm volatile("global_load_async_to_lds_b128 %0, %1, off"
                 :: "v"(l00), "v"(g00) : "memory");
    asm volatile("global_load_async_to_lds_b128 %0, %1, off"
                 :: "v"(l01), "v"(g01) : "memory");
    asm volatile("global_load_async_to_lds_b128 %0, %1, off"
                 :: "v"(l10), "v"(g10) : "memory");
    asm volatile("global_load_async_to_lds_b128 %0, %1, off"
                 :: "v"(l11), "v"(g11) : "memory");
    // Each lane reads back only its own LDS slot -> no barrier, just the
    // async counter drain.
    asm volatile("s_wait_asynccnt 0x0" ::: "memory");

    const v4f p00 = *(const v4f*)&taps[0 * CH + co];
    const v4f p01 = *(const v4f*)&taps[1 * CH + co];
    const v4f p10 = *(const v4f*)&taps[2 * CH + co];
    const v4f p11 = *(const v4f*)&taps[3 * CH + co];

    // Match reference FP association: row lerp (fy) first, then column lerp (fx).
    const float wy0 = 1.0f - fy, wx0 = 1.0f - fx;
    const v4f ra = p00 * wy0 + p10 * fy;   // rows[:, x+x0]
    const v4f rb = p01 * wy0 + p11 * fy;   // rows[:, x+x1]
    const v4f o  = ra * wx0 + rb * fx;

    // Streaming output (30 MB, zero reuse): non-temporal 128-bit store.
    const size_t cell = ((size_t)roi * POOL + ph) * POOL + pw;
    __builtin_nontemporal_store(o, (v4f*)(out + cell * CH + co));
}

extern "C" void kernel_launch(void* const* d_in, const int* in_sizes, int n_in,
                              void* d_out, int out_size, void* d_ws, size_t ws_size,
                              hipStream_t stream) {
    (void)in_sizes; (void)n_in; (void)out_size; (void)d_ws; (void)ws_size;
    const float* img  = (const float*)d_in[0];   // (1,200,200,512) f32
    const float* rois = (const float*)d_in[1];   // (1,300,4) f32
    float*       out  = (float*)d_out;           // (1,300,7,7,512) f32

    dim3 grid(POOL, POOL, N_ROI);                // (pw, ph, roi)
    roi_pool_bilinear_async<<<grid, 128, 0, stream>>>(img, rois, out);
}